// PredictionDecoder_56891136802987
// MI455X (gfx1250) — compile-verified
//
#include <hip/hip_runtime.h>
#include <cstdint>

#define NUM_B   8
#define NUM_A   196416
#define NUM_C   6
#define PRE_K   1000
#define PADK    1024
#define MAXPC   100
#define MAXD    100
// logit(0.05) = ln(0.05/0.95)
#define LOGIT_THR (-2.9444389791664403f)

typedef float v2f __attribute__((ext_vector_type(2)));
typedef float v8f __attribute__((ext_vector_type(8)));

// gcc-style int4 to match the async-to-LDS builtin's parameter type exactly
typedef int v4i __attribute__((vector_size(16)));
typedef __attribute__((address_space(1))) v4i global_v4i;
typedef __attribute__((address_space(3))) v4i shared_v4i;

// ---------- helpers ----------

// monotone float -> uint key (bigger key == bigger float)
__device__ __forceinline__ unsigned fkey(float f) {
  unsigned u = __float_as_uint(f);
  return (u & 0x80000000u) ? ~u : (u | 0x80000000u);
}
__device__ __forceinline__ float ikey(unsigned k) {
  unsigned u = (k & 0x80000000u) ? (k & 0x7FFFFFFFu) : ~k;
  return __uint_as_float(u);
}

__device__ __forceinline__ unsigned ballot32(bool b) {
#if __has_builtin(__builtin_amdgcn_ballot_w32)
  return __builtin_amdgcn_ballot_w32(b);
#else
  return (unsigned)__ballot(b);
#endif
}

__device__ __forceinline__ void wait_async0() {
#if __has_builtin(__builtin_amdgcn_s_wait_asynccnt)
  __builtin_amdgcn_s_wait_asynccnt(0);
#else
  asm volatile("s_wait_asynccnt 0" ::: "memory");
#endif
}

// keras.io RetinaNet AnchorBox, H=W=1024: levels 3..7, ratios outer x scales inner
__device__ __forceinline__ void anchor_at(int a, float& cx, float& cy, float& aw, float& ah) {
  int base, fw, stride; float area;
  if (a < 147456)      { base = 0;      fw = 128; stride = 8;   area = 1024.f;   }
  else if (a < 184320) { base = 147456; fw = 64;  stride = 16;  area = 4096.f;   }
  else if (a < 193536) { base = 184320; fw = 32;  stride = 32;  area = 16384.f;  }
  else if (a < 195840) { base = 193536; fw = 16;  stride = 64;  area = 65536.f;  }
  else                 { base = 195840; fw = 8;   stride = 128; area = 262144.f; }
  int rem  = a - base;
  int k    = rem % 9;
  int cell = rem / 9;
  int x = cell % fw, y = cell / fw;
  int r = k / 3, s = k % 3;
  float ratio = (r == 0) ? 0.5f : ((r == 1) ? 1.0f : 2.0f);
  float scale = (s == 0) ? 1.0f : ((s == 1) ? 1.2599210499f : 1.5874010520f);
  float h = sqrtf(area / ratio);
  float w = area / h;
  aw = w * scale; ah = h * scale;
  cx = ((float)x + 0.5f) * (float)stride;
  cy = ((float)y + 0.5f) * (float)stride;
}

// ---------- kernel 1: fused anchor decode + logit transpose ----------
__global__ void __launch_bounds__(256) decode_kernel(const float* __restrict__ pred,
                                                     float* __restrict__ boxes,
                                                     float* __restrict__ logits) {
  int g = blockIdx.x * blockDim.x + threadIdx.x;
  if (g >= NUM_B * NUM_A) return;
  int b = g / NUM_A;
  int a = g - b * NUM_A;

  float cx, cy, aw, ah;
  anchor_at(a, cx, cy, aw, ah);

  const float* p = pred + (size_t)g * 10;
  __builtin_prefetch(p + 2560, 0, 1);   // -> global_prefetch_b8

  float bx = p[0] * 0.1f, by = p[1] * 0.1f;
  float bw = p[2] * 0.2f, bh = p[3] * 0.2f;
  float cxp = bx * aw + cx;
  float cyp = by * ah + cy;
  float wp = expf(bw) * aw * 0.5f;
  float hp = expf(bh) * ah * 0.5f;

  float* ob = boxes + (size_t)g * 4;
  ob[0] = cxp - wp; ob[1] = cyp - hp; ob[2] = cxp + wp; ob[3] = cyp + hp;

#pragma unroll
  for (int c = 0; c < NUM_C; ++c)
    logits[(size_t)(b * NUM_C + c) * NUM_A + a] = p[4 + c];
}

// ---------- kernel 2: per-(batch,class) top-1000 select + NMS ----------
// dynamic LDS layout (bytes)
#define SM_SUP   0        // u16[1024*64]  suppression bit matrix   (128 KB)
#define SM_KEY   131072   // u32[1024]
#define SM_IDX   135168   // i32[1024]
#define SM_BOX   139264   // float4[1024]
#define SM_AREA  155648   // f32[1024]
#define SM_KEEP  159744   // u32[32]
#define SM_HIST  159872   // u32[256]
#define SM_MISC  160896   // u32[8]
#define SMEM_BYTES 160928

__global__ void __launch_bounds__(1024) select_nms_kernel(const float* __restrict__ logits,
                                                          const float* __restrict__ boxes,
                                                          float* __restrict__ cls_scores,
                                                          float* __restrict__ cls_boxes) {
  extern __shared__ unsigned char smem[];
  unsigned short* sup16 = (unsigned short*)(smem + SM_SUP);
  unsigned*       key_s = (unsigned*)(smem + SM_KEY);
  int*            idx_s = (int*)(smem + SM_IDX);
  float4*         box_s = (float4*)(smem + SM_BOX);
  float*          area_s = (float*)(smem + SM_AREA);
  unsigned*       keep_s = (unsigned*)(smem + SM_KEEP);
  unsigned*       hist = (unsigned*)(smem + SM_HIST);
  unsigned*       miscu = (unsigned*)(smem + SM_MISC);

  const int tid  = threadIdx.x;
  const int nthr = 1024;
  const int blk  = blockIdx.x;           // b*NUM_C + c
  const int b    = blk / NUM_C;
  const float* lp = logits + (size_t)blk * NUM_A;

  // ---- Phase A: 4x8-bit radix select of the PRE_K-th largest key ----
  unsigned prefix = 0, pmask = 0;
  int want = PRE_K, totalG = 0;
  for (int pass = 0; pass < 4; ++pass) {
    int shift = 24 - 8 * pass;
    for (int i = tid; i < 256; i += nthr) hist[i] = 0;
    __syncthreads();
    for (int i = tid; i < NUM_A; i += nthr) {
      __builtin_prefetch(lp + i + 4 * nthr, 0, 1);
      unsigned u = fkey(lp[i]);
      if ((u & pmask) == prefix) atomicAdd(&hist[(u >> shift) & 255u], 1u);
    }
    __syncthreads();
    if (tid == 0) {
      unsigned cum = 0, sel = 0;
      for (int d = 255; d >= 0; --d) {
        unsigned h = hist[d];
        if (cum + h >= (unsigned)want) { sel = (unsigned)d; break; }
        cum += h;
      }
      miscu[0] = sel; miscu[1] = cum;
    }
    __syncthreads();
    unsigned sel = miscu[0], cum = miscu[1];
    prefix |= sel << shift;
    pmask  |= 0xFFu << shift;
    want   -= (int)cum;
    totalG += (int)cum;
    __syncthreads();
  }
  const unsigned pivot = prefix;   // exact key of the 1000th largest
  const int G = totalG;            // #keys strictly greater than pivot (<1000)

  // ---- Phase B: compact top-PRE_K (strictly greater, then ties) ----
  if (tid == 0) { miscu[2] = 0; miscu[3] = 0; }
  __syncthreads();
  for (int i = tid; i < NUM_A; i += nthr) {
    unsigned u = fkey(lp[i]);
    if (u > pivot) { unsigned p = atomicAdd(&miscu[2], 1u); key_s[p] = u; idx_s[p] = i; }
  }
  __syncthreads();
  for (int i = tid; i < NUM_A; i += nthr) {
    unsigned u = fkey(lp[i]);
    if (u == pivot) {
      unsigned e = atomicAdd(&miscu[3], 1u);
      unsigned p = (unsigned)G + e;
      if (p < PRE_K) { key_s[p] = u; idx_s[p] = i; }
    }
  }
  __syncthreads();
  for (int i = PRE_K + tid; i < PADK; i += nthr) { key_s[i] = 0u; idx_s[i] = -1; }
  __syncthreads();

  // ---- Phase C: bitonic sort 1024 (descending) keys + index payload ----
  for (int k2 = 2; k2 <= PADK; k2 <<= 1) {
    for (int j = k2 >> 1; j > 0; j >>= 1) {
      int i = tid, ixj = i ^ j;
      if (ixj > i) {
        unsigned ki = key_s[i], kj = key_s[ixj];
        bool descB = (i & k2) == 0;
        if (descB ? (ki < kj) : (ki > kj)) {
          key_s[i] = kj; key_s[ixj] = ki;
          int t = idx_s[i]; idx_s[i] = idx_s[ixj]; idx_s[ixj] = t;
        }
      }
      __syncthreads();
    }
  }

  // ---- Phase D: gather candidate boxes into LDS (async b128), areas ----
  {
    int gi = idx_s[tid];
    bool valid = (tid < PRE_K) && (gi >= 0);
    if (valid) {
      const float* src = boxes + ((size_t)b * NUM_A + (size_t)gi) * 4;
#if __has_builtin(__builtin_amdgcn_global_load_async_to_lds_b128)
      __builtin_amdgcn_global_load_async_to_lds_b128(
          (global_v4i*)(const_cast<float*>(src)),
          (shared_v4i*)(&box_s[tid]), 0, 0);
#else
      box_s[tid] = make_float4(src[0], src[1], src[2], src[3]);
#endif
    } else {
      box_s[tid] = make_float4(0.f, 0.f, 0.f, 0.f);
    }
#if __has_builtin(__builtin_amdgcn_global_load_async_to_lds_b128)
    wait_async0();
#endif
    __syncthreads();
    float4 bb = box_s[tid];
    area_s[tid] = (bb.z - bb.x) * (bb.w - bb.y);
  }

  // keep0 = score > CONF_THR  (strict; monotone in key space)
  const unsigned THRKEY = fkey(LOGIT_THR);
  {
    bool kb = (tid < PRE_K) && (idx_s[tid] >= 0) && (key_s[tid] > THRKEY);
    __syncthreads();
    unsigned bal = ballot32(kb);
    if ((tid & 31) == 0) keep_s[tid >> 5] = bal;
  }
  __syncthreads();

  // ---- Phase E: 1024x1024 suppression bit-matrix, 16x16 WMMA tiles ----
  // denom tile: area_i + area_j via V_WMMA_F32_16X16X4_F32 (rank-2 matmul)
  {
    const int wave = tid >> 5;
    const int lane = tid & 31;
    const bool lo  = lane < 16;
    const int l16  = lane & 15;
    for (int t = wave; t < 64 * 64; t += 32) {   // wave-uniform bounds: EXEC all-1 at WMMA
      int tr = t >> 6, tc = t & 63;
#if __has_builtin(__builtin_amdgcn_wmma_f32_16x16x4_f32)
      // A (16x4): K0 = area_row, K1 = 1, K2/K3 = 0 ; B (4x16): K0 = 1, K1 = area_col
      v2f af, bf;
      af.x = lo ? area_s[tr * 16 + l16] : 0.f;   // lanes0-15:K0, lanes16-31:K2
      af.y = lo ? 1.f : 0.f;                     // lanes0-15:K1, lanes16-31:K3
      bf.x = lo ? 1.f : 0.f;                     // row K0 / row K2
      bf.y = lo ? area_s[tc * 16 + l16] : 0.f;   // row K1 / row K3
      v8f cacc = {};
      v8f dfrag = __builtin_amdgcn_wmma_f32_16x16x4_f32(
          false, af, false, bf, (short)0, cacc, false, false);
#endif
#pragma unroll
      for (int v = 0; v < 8; ++v) {
        int M = lo ? v : v + 8;                  // C/D layout: VGPR v -> rows v, v+8
        int r  = tr * 16 + M;
        int cI = tc * 16 + l16;
        float4 br = box_s[r], bc = box_s[cI];
        float iw = fminf(br.z, bc.z) - fmaxf(br.x, bc.x);
        float ih = fminf(br.w, bc.w) - fmaxf(br.y, bc.y);
        float inter = fmaxf(iw, 0.f) * fmaxf(ih, 0.f);
#if __has_builtin(__builtin_amdgcn_wmma_f32_16x16x4_f32)
        float S = dfrag[v];
#else
        float S = area_s[r] + area_s[cI];
#endif
        // iou > 0.5  <=>  3*inter > S + eps   (no division)
        bool bit = (3.f * inter) > (S + 1e-8f);
        unsigned m = ballot32(bit);              // low16: row tr*16+v, high16: row +8
        if (lane == 0) {
          sup16[(tr * 16 + v) * 64 + tc]     = (unsigned short)(m & 0xFFFFu);
          sup16[(tr * 16 + v + 8) * 64 + tc] = (unsigned short)(m >> 16);
        }
      }
    }
  }
  __syncthreads();

  // ---- Phase F: serial greedy sweep, one wave, lane w owns keep word w ----
  if (tid < 32) {
    volatile unsigned* kp = keep_s;
    const unsigned* sup32 = (const unsigned*)sup16;   // u16 pairs == u32 words (LE)
    for (int i = 0; i < PRE_K; ++i) {
      unsigned kw = kp[i >> 5];
      if ((kw >> (i & 31)) & 1u) {
        unsigned m = sup32[i * 32 + tid];
        int wi = i >> 5;
        if (tid < wi)       m = 0;
        else if (tid == wi) m &= ~((2u << (i & 31)) - 1u);  // only j > i suppress
        kp[tid] = kp[tid] & ~m;
      }
    }
  }
  __syncthreads();

  // ---- Phase G: emit per-class top-100 (already score-sorted) ----
  if (tid == 0) {
    float* os = cls_scores + (size_t)blk * MAXPC;
    float* ob = cls_boxes + (size_t)blk * MAXPC * 4;
    int cnt = 0;
    for (int i = 0; i < PRE_K && cnt < MAXPC; ++i) {
      if ((keep_s[i >> 5] >> (i & 31)) & 1u) {
        float lg = ikey(key_s[i]);
        float sc = 1.f / (1.f + expf(-lg));
        float4 bb = box_s[i];
        os[cnt] = sc;
        ob[cnt * 4 + 0] = bb.x; ob[cnt * 4 + 1] = bb.y;
        ob[cnt * 4 + 2] = bb.z; ob[cnt * 4 + 3] = bb.w;
        ++cnt;
      }
    }
    for (; cnt < MAXPC; ++cnt) {
      os[cnt] = -1.f;
      ob[cnt * 4 + 0] = 0.f; ob[cnt * 4 + 1] = 0.f;
      ob[cnt * 4 + 2] = 0.f; ob[cnt * 4 + 3] = 0.f;
    }
  }
}

// ---------- kernel 3: per-batch cross-class merge to top-100 ----------
__global__ void __launch_bounds__(1024) merge_kernel(const float* __restrict__ cls_scores,
                                                     const float* __restrict__ cls_boxes,
                                                     float* __restrict__ out) {
  __shared__ float sc_s[1024];
  __shared__ int   pl_s[1024];
  const int b = blockIdx.x, tid = threadIdx.x;

  float s = -2.f;
  if (tid < NUM_C * MAXPC) s = cls_scores[(size_t)b * NUM_C * MAXPC + tid];
  sc_s[tid] = s;
  pl_s[tid] = tid;
  __syncthreads();

  for (int k2 = 2; k2 <= 1024; k2 <<= 1) {
    for (int j = k2 >> 1; j > 0; j >>= 1) {
      int i = tid, ixj = i ^ j;
      if (ixj > i) {
        float ki = sc_s[i], kj = sc_s[ixj];
        bool descB = (i & k2) == 0;
        if (descB ? (ki < kj) : (ki > kj)) {
          sc_s[i] = kj; sc_s[ixj] = ki;
          int t = pl_s[i]; pl_s[i] = pl_s[ixj]; pl_s[ixj] = t;
        }
      }
      __syncthreads();
    }
  }

  float* oB = out;                                     // [8,100,4]
  float* oS = out + (size_t)NUM_B * MAXD * 4;          // [8,100]
  float* oC = oS + (size_t)NUM_B * MAXD;               // [8,100]
  if (tid < MAXD) {
    float sv = sc_s[tid];
    int t = pl_s[tid];
    bool valid = sv > -0.5f;
    const float* bp = cls_boxes + ((size_t)b * NUM_C * MAXPC + t) * 4;
    float x1 = valid ? bp[0] : 0.f;
    float y1 = valid ? bp[1] : 0.f;
    float x2 = valid ? bp[2] : 0.f;
    float y2 = valid ? bp[3] : 0.f;
    size_t o = (size_t)b * MAXD + tid;
    oB[o * 4 + 0] = x1; oB[o * 4 + 1] = y1; oB[o * 4 + 2] = x2; oB[o * 4 + 3] = y2;
    oS[o] = valid ? sv : 0.f;
    oC[o] = valid ? (float)(t / MAXPC) : 0.f;
  }
  __syncthreads();
  if (tid == 0) {
    int cnt = 0;
    for (int i = 0; i < MAXD; ++i)
      if (sc_s[i] > -0.5f) ++cnt;
    ((int*)out)[NUM_B * MAXD * 4 + 2 * NUM_B * MAXD + b] = cnt;   // valid[b], int32
  }
}

// ---------- launch ----------
extern "C" void kernel_launch(void* const* d_in, const int* in_sizes, int n_in,
                              void* d_out, int out_size, void* d_ws, size_t ws_size,
                              hipStream_t stream) {
  (void)in_sizes; (void)n_in; (void)out_size; (void)ws_size;
  const float* preds = (const float*)d_in[1];   // d_in[0]=images: only H=W=1024 used

  float* ws = (float*)d_ws;
  float* boxes_ws      = ws;                                            // B*A*4
  float* logits_ws     = boxes_ws + (size_t)NUM_B * NUM_A * 4;          // B*C*A
  float* cls_scores_ws = logits_ws + (size_t)NUM_B * NUM_C * NUM_A;     // B*C*100
  float* cls_boxes_ws  = cls_scores_ws + (size_t)NUM_B * NUM_C * MAXPC; // B*C*100*4

  (void)hipFuncSetAttribute((const void*)select_nms_kernel,
                            hipFuncAttributeMaxDynamicSharedMemorySize, SMEM_BYTES);

  int total = NUM_B * NUM_A;
  decode_kernel<<<(total + 255) / 256, 256, 0, stream>>>(preds, boxes_ws, logits_ws);
  select_nms_kernel<<<NUM_B * NUM_C, 1024, SMEM_BYTES, stream>>>(
      logits_ws, boxes_ws, cls_scores_ws, cls_boxes_ws);
  merge_kernel<<<NUM_B, 1024, 0, stream>>>(cls_scores_ws, cls_boxes_ws, (float*)d_out);
}